// LLaMACausalSelfAttention_8796093022663
// MI455X (gfx1250) — compile-verified
//
#include <hip/hip_runtime.h>
#include <hip/hip_bf16.h>

// Problem constants (match reference)
#define B_    32
#define S_    2048
#define D_    4096
#define H_    32
#define KVH_  8
#define HD_   128
#define HALF_ 64

// Workspace layout (floats)
#define QOFF_ 0                       // q:   32 x 4096
#define KOFF_ (B_ * D_)               // k:   32 x 1024
#define VOFF_ (KOFF_ + B_ * KVH_ * HD_)
#define AOFF_ (VOFF_ + B_ * KVH_ * HD_) // attn_out: 32 x 4096

typedef __attribute__((ext_vector_type(2))) float v2f;
typedef __attribute__((ext_vector_type(8))) float v8f;

__device__ __forceinline__ v8f zero8() {
    v8f z;
#pragma unroll
    for (int i = 0; i < 8; ++i) z[i] = 0.0f;
    return z;
}

// C[0:32][n0:n0+32] = X[0:32][0:K] * W[0:K][n0:n0+32]
// block = 256 threads (8 wave32). Split-K across waves, deterministic LDS reduce.
// A-frag layout per ISA 7.12.2 (f32 16x4): lanes 0-15 hold K={k,k+1}, lanes 16-31 K={k+2,k+3}.
// B-frag assumed symmetric (4x16: v0 = rows k / k+2 across lane halves, v1 = rows k+1 / k+3).
__device__ __forceinline__ void gemm_rows32_tile(
    const float* __restrict__ X, int Kdim,
    const float* __restrict__ W, int Nld, int n0,
    float* __restrict__ C, int Cld)
{
    const int tid  = threadIdx.x;
    const int wave = tid >> 5;
    const int lane = tid & 31;
    const int l16  = lane & 15;
    const int koff = (lane >> 4) << 1;   // 0 or 2

    const int kslice = Kdim >> 3;        // 8 waves split K
    const int kbeg   = wave * kslice;
    const int kend   = kbeg + kslice;

    v8f c00 = zero8(), c01 = zero8(), c10 = zero8(), c11 = zero8();

    const float* xr0 = X + (size_t)l16 * Kdim;         // M-tile 0 row
    const float* xr1 = X + (size_t)(l16 + 16) * Kdim;  // M-tile 1 row
    const float* wc0 = W + n0 + l16;                   // N-tile 0 column
    const float* wc1 = W + n0 + 16 + l16;              // N-tile 1 column

#pragma unroll 2
    for (int k = kbeg; k < kend; k += 4) {
        const int kk = k + koff;
        v2f a0, a1, b0, b1;
        a0.x = xr0[kk];  a0.y = xr0[kk + 1];
        a1.x = xr1[kk];  a1.y = xr1[kk + 1];
        b0.x = wc0[(size_t)kk * Nld]; b0.y = wc0[(size_t)(kk + 1) * Nld];
        b1.x = wc1[(size_t)kk * Nld]; b1.y = wc1[(size_t)(kk + 1) * Nld];
        c00 = __builtin_amdgcn_wmma_f32_16x16x4_f32(false, a0, false, b0, (short)0, c00, false, false);
        c01 = __builtin_amdgcn_wmma_f32_16x16x4_f32(false, a0, false, b1, (short)0, c01, false, false);
        c10 = __builtin_amdgcn_wmma_f32_16x16x4_f32(false, a1, false, b0, (short)0, c10, false, false);
        c11 = __builtin_amdgcn_wmma_f32_16x16x4_f32(false, a1, false, b1, (short)0, c11, false, false);
    }

    // Reduce 8 partial 32x32 tiles (fixed order -> deterministic)
    __shared__ float part[8][1024];
#pragma unroll
    for (int r = 0; r < 8; ++r) {
        part[wave][0 * 256 + r * 32 + lane] = c00[r];
        part[wave][1 * 256 + r * 32 + lane] = c01[r];
        part[wave][2 * 256 + r * 32 + lane] = c10[r];
        part[wave][3 * 256 + r * 32 + lane] = c11[r];
    }
    __syncthreads();

#pragma unroll
    for (int i = 0; i < 4; ++i) {
        const int p = tid + i * 256;
        float s = 0.0f;
#pragma unroll
        for (int w = 0; w < 8; ++w) s += part[w][p];
        const int mtile = p >> 9;
        const int ntile = (p >> 8) & 1;
        const int r     = (p >> 5) & 7;
        const int L     = p & 31;
        const int M     = mtile * 16 + r + ((L >> 4) << 3);
        const int Ncol  = n0 + ntile * 16 + (L & 15);
        C[(size_t)M * Cld + Ncol] = s;
    }
}

// Fused QKV projection: blocks [0,128) -> q, [128,160) -> k, [160,192) -> v
__global__ void __launch_bounds__(256) qkv_gemm_kernel(
    const float* __restrict__ x, const float* __restrict__ wq,
    const float* __restrict__ wk, const float* __restrict__ wv,
    float* __restrict__ ws)
{
    float* q_ws = ws + QOFF_;
    float* k_ws = ws + KOFF_;
    float* v_ws = ws + VOFF_;
    const int T = blockIdx.x;
    if (T < 128) {
        gemm_rows32_tile(x, D_, wq, D_, T * 32, q_ws, D_);
    } else if (T < 160) {
        gemm_rows32_tile(x, D_, wk, KVH_ * HD_, (T - 128) * 32, k_ws, KVH_ * HD_);
    } else {
        gemm_rows32_tile(x, D_, wv, KVH_ * HD_, (T - 160) * 32, v_ws, KVH_ * HD_);
    }
}

// RoPE at start_pos on q (32 heads) and new k (8 heads), in workspace.
__global__ void __launch_bounds__(256) rope_kernel(
    float* __restrict__ ws, const float* __restrict__ fcos,
    const float* __restrict__ fsin, const int* __restrict__ spp)
{
    const int idx = blockIdx.x * blockDim.x + threadIdx.x;
    if (idx >= B_ * (H_ + KVH_) * HALF_) return;
    const int sp   = *spp;
    const int p    = idx & (HALF_ - 1);
    const int head = (idx >> 6) % (H_ + KVH_);
    const int b    = idx / ((H_ + KVH_) * HALF_);
    const float fc = fcos[(size_t)sp * HALF_ + p];
    const float fs = fsin[(size_t)sp * HALF_ + p];
    float* base = (head < H_)
        ? (ws + QOFF_ + ((size_t)b * H_ + head) * HD_)
        : (ws + KOFF_ + ((size_t)b * KVH_ + (head - H_)) * HD_);
    const float r = base[2 * p];
    const float i = base[2 * p + 1];
    base[2 * p]     = r * fc - i * fs;
    base[2 * p + 1] = r * fs + i * fc;
}

// One block per (kv-head g, batch b): 4 q-heads share the K/V stream (read once).
__global__ void __launch_bounds__(256) attn_kernel(
    float* __restrict__ ws, const float* __restrict__ k_cache,
    const float* __restrict__ v_cache, const int* __restrict__ spp)
{
    const float* q_ws = ws + QOFF_;
    const float* k_ws = ws + KOFF_;
    const float* v_ws = ws + VOFF_;
    float* attn_out   = ws + AOFF_;

    const int g   = blockIdx.x;
    const int b   = blockIdx.y;
    const int sp  = *spp;
    const int L   = sp + 1;     // valid positions 0..sp; s==sp uses new k/v
    const int tid = threadIdx.x;

    __shared__ float q_sm[4][HD_];
    __shared__ float sc[4][S_];
    __shared__ float red[4][256];
    __shared__ float parts[2][4][HD_];

    for (int i = tid; i < 4 * HD_; i += 256) {
        const int hh = i >> 7, d = i & (HD_ - 1);
        q_sm[hh][d] = q_ws[((size_t)b * H_ + g * 4 + hh) * HD_ + d];
    }
    __syncthreads();

    const float scale = 0.08838834764831845f;   // 1/sqrt(128)
    float mx0 = -3e38f, mx1 = -3e38f, mx2 = -3e38f, mx3 = -3e38f;
    for (int s = tid; s < L; s += 256) {
        const float* kv = (s == sp)
            ? (k_ws + ((size_t)b * KVH_ + g) * HD_)
            : (k_cache + (((size_t)b * S_ + s) * KVH_ + g) * HD_);
        float a0 = 0.f, a1 = 0.f, a2 = 0.f, a3 = 0.f;
#pragma unroll
        for (int d = 0; d < HD_; d += 4) {
            const float4 kk = *(const float4*)(kv + d);
            a0 += q_sm[0][d]*kk.x + q_sm[0][d+1]*kk.y + q_sm[0][d+2]*kk.z + q_sm[0][d+3]*kk.w;
            a1 += q_sm[1][d]*kk.x + q_sm[1][d+1]*kk.y + q_sm[1][d+2]*kk.z + q_sm[1][d+3]*kk.w;
            a2 += q_sm[2][d]*kk.x + q_sm[2][d+1]*kk.y + q_sm[2][d+2]*kk.z + q_sm[2][d+3]*kk.w;
            a3 += q_sm[3][d]*kk.x + q_sm[3][d+1]*kk.y + q_sm[3][d+2]*kk.z + q_sm[3][d+3]*kk.w;
        }
        a0 *= scale; a1 *= scale; a2 *= scale; a3 *= scale;
        sc[0][s] = a0; sc[1][s] = a1; sc[2][s] = a2; sc[3][s] = a3;
        mx0 = fmaxf(mx0, a0); mx1 = fmaxf(mx1, a1);
        mx2 = fmaxf(mx2, a2); mx3 = fmaxf(mx3, a3);
    }
    red[0][tid] = mx0; red[1][tid] = mx1; red[2][tid] = mx2; red[3][tid] = mx3;
    __syncthreads();
    for (int off = 128; off > 0; off >>= 1) {
        if (tid < off) {
#pragma unroll
            for (int hh = 0; hh < 4; ++hh)
                red[hh][tid] = fmaxf(red[hh][tid], red[hh][tid + off]);
        }
        __syncthreads();
    }
    const float m0 = red[0][0], m1 = red[1][0], m2 = red[2][0], m3 = red[3][0];
    __syncthreads();

    float s0 = 0.f, s1 = 0.f, s2 = 0.f, s3 = 0.f;
    for (int s = tid; s < L; s += 256) {
        const float w0 = __expf(sc[0][s] - m0); sc[0][s] = w0; s0 += w0;
        const float w1 = __expf(sc[1][s] - m1); sc[1][s] = w1; s1 += w1;
        const float w2 = __expf(sc[2][s] - m2); sc[2][s] = w2; s2 += w2;
        const float w3 = __expf(sc[3][s] - m3); sc[3][s] = w3; s3 += w3;
    }
    red[0][tid] = s0; red[1][tid] = s1; red[2][tid] = s2; red[3][tid] = s3;
    __syncthreads();
    for (int off = 128; off > 0; off >>= 1) {
        if (tid < off) {
#pragma unroll
            for (int hh = 0; hh < 4; ++hh)
                red[hh][tid] += red[hh][tid + off];
        }
        __syncthreads();
    }
    const float inv0 = 1.0f / red[0][0], inv1 = 1.0f / red[1][0];
    const float inv2 = 1.0f / red[2][0], inv3 = 1.0f / red[3][0];

    // Pass 2: thread = output dim d; two s-groups for memory-level parallelism.
    const int d  = tid & (HD_ - 1);
    const int sg = tid >> 7;
    float a0 = 0.f, a1 = 0.f, a2 = 0.f, a3 = 0.f;
    const float* vbase = v_cache + ((size_t)b * S_ * KVH_ + g) * HD_ + d;
#pragma unroll 4
    for (int s = sg; s < sp; s += 2) {
        const float v = vbase[(size_t)s * (KVH_ * HD_)];
        a0 += sc[0][s] * v; a1 += sc[1][s] * v;
        a2 += sc[2][s] * v; a3 += sc[3][s] * v;
    }
    if (sg == 0) {  // inject new token's v at s == sp
        const float v = v_ws[((size_t)b * KVH_ + g) * HD_ + d];
        a0 += sc[0][sp] * v; a1 += sc[1][sp] * v;
        a2 += sc[2][sp] * v; a3 += sc[3][sp] * v;
    }
    parts[sg][0][d] = a0; parts[sg][1][d] = a1;
    parts[sg][2][d] = a2; parts[sg][3][d] = a3;
    __syncthreads();
    if (sg == 0) {
        float* outp = attn_out + (size_t)b * D_ + (size_t)(g * 4) * HD_ + d;
        outp[0 * HD_] = (parts[0][0][d] + parts[1][0][d]) * inv0;
        outp[1 * HD_] = (parts[0][1][d] + parts[1][1][d]) * inv1;
        outp[2 * HD_] = (parts[0][2][d] + parts[1][2][d]) * inv2;
        outp[3 * HD_] = (parts[0][3][d] + parts[1][3][d]) * inv3;
    }
}

__global__ void __launch_bounds__(256) out_gemm_kernel(
    const float* __restrict__ ws, const float* __restrict__ wo,
    float* __restrict__ out)
{
    gemm_rows32_tile(ws + AOFF_, D_, wo, D_, blockIdx.x * 32, out, D_);
}

extern "C" void kernel_launch(void* const* d_in, const int* in_sizes, int n_in,
                              void* d_out, int out_size, void* d_ws, size_t ws_size,
                              hipStream_t stream) {
    const float* x  = (const float*)d_in[0];
    const float* wq = (const float*)d_in[1];
    const float* wk = (const float*)d_in[2];
    const float* wv = (const float*)d_in[3];
    const float* wo = (const float*)d_in[4];
    const float* kc = (const float*)d_in[5];
    const float* vc = (const float*)d_in[6];
    const float* fc = (const float*)d_in[7];
    const float* fs = (const float*)d_in[8];
    // d_in[9] = mask: unused, validity derived from start_pos
    const int* sp   = (const int*)d_in[10];
    float* out = (float*)d_out;
    float* ws  = (float*)d_ws;

    qkv_gemm_kernel<<<192, 256, 0, stream>>>(x, wq, wk, wv, ws);
    rope_kernel<<<(B_ * (H_ + KVH_) * HALF_ + 255) / 256, 256, 0, stream>>>(ws, fc, fs, sp);
    attn_kernel<<<dim3(KVH_, B_), 256, 0, stream>>>(ws, kc, vc, sp);
    out_gemm_kernel<<<128, 256, 0, stream>>>(ws, wo, out);
}